// UncertaintyEstimator_45749991637200
// MI455X (gfx1250) — compile-verified
//
#include <hip/hip_runtime.h>
#include <math.h>

#define BINS 50

typedef __attribute__((ext_vector_type(2))) float v2f;
typedef __attribute__((ext_vector_type(8))) float v8f;

#define NB_COV  512
#define NB_SMM  1024
#define NB_ENTJ 1024
#define NB_ENTM 64
#define NB_HISTJ 2048

// ---------------------------------------------------------------- zero ws
__global__ void zero_ws_kernel(unsigned int* __restrict__ ws, long nWords) {
  long i = (long)blockIdx.x * blockDim.x + threadIdx.x;
  long stride = (long)gridDim.x * blockDim.x;
  for (; i < nWords; i += stride) ws[i] = 0u;
}

// ------------------------------------------- Gram (cov) via v_wmma_f32_16x16x4_f32
// Features per row: [q0 q1 r0 r1 1] (the constant-1 feature makes row/col 4 of the
// Gram the column sums + count). A(16x4) and B(4x16) per-lane layouts coincide for
// A = B^T = X chunk, so one register pair feeds both WMMA operands.
__global__ void __launch_bounds__(32) cov_wmma_kernel(
    const float* __restrict__ q, const float* __restrict__ r,
    float* __restrict__ gramPart, float* __restrict__ mmPart, int nChunks) {
  const int lane = threadIdx.x;
  const int m = lane & 15;   // feature index (M for A, N for B)
  const int h = lane >> 4;   // K half: lanes 0-15 -> K=0,1 ; lanes 16-31 -> K=2,3
  v8f acc = {0.f, 0.f, 0.f, 0.f, 0.f, 0.f, 0.f, 0.f};
  float qmn = __builtin_inff(), qmx = -__builtin_inff();
  float rmn = __builtin_inff(), rmx = -__builtin_inff();

  for (int c = blockIdx.x; c < nChunks; c += gridDim.x) {   // uniform per wave
    int row0 = c * 4 + 2 * h;
    float x0, x1;
    if (m < 2) {
      x0 = q[row0 * 2 + m];
      x1 = q[(row0 + 1) * 2 + m];
      qmn = fminf(qmn, fminf(x0, x1));
      qmx = fmaxf(qmx, fmaxf(x0, x1));
    } else if (m < 4) {
      x0 = r[row0 * 2 + (m - 2)];
      x1 = r[(row0 + 1) * 2 + (m - 2)];
      rmn = fminf(rmn, fminf(x0, x1));
      rmx = fmaxf(rmx, fmaxf(x0, x1));
    } else if (m == 4) {
      x0 = 1.f; x1 = 1.f;
    } else {
      x0 = 0.f; x1 = 0.f;
    }
    v2f a; a.x = x0; a.y = x1;
    // D = A*B + C ; A == B^T == X-chunk -> D accumulates X^T X (5x5 live block)
    acc = __builtin_amdgcn_wmma_f32_16x16x4_f32(false, a, false, a, (short)0, acc,
                                                false, false);
  }

  // C/D layout: vgpr v, lanes 0-15 hold (M=v, N=lane). We need M,N in 0..4.
  if (lane < 5) {
#pragma unroll
    for (int v = 0; v < 5; ++v)
      gramPart[blockIdx.x * 32 + lane * 5 + v] = acc[v]; // t = n*5 + m -> G[m][n]
  }
  // wave-wide min/max reduce (wave32)
#pragma unroll
  for (int o = 16; o; o >>= 1) {
    qmn = fminf(qmn, __shfl_xor(qmn, o, 32));
    qmx = fmaxf(qmx, __shfl_xor(qmx, o, 32));
    rmn = fminf(rmn, __shfl_xor(rmn, o, 32));
    rmx = fmaxf(rmx, __shfl_xor(rmx, o, 32));
  }
  if (lane == 0) {
    float* p = mmPart + blockIdx.x * 4;
    p[0] = qmn; p[1] = qmx; p[2] = rmn; p[3] = rmx;
  }
}

// --------------------------- combine partial Grams, mean, cov, 4x4 Cholesky
__global__ void __launch_bounds__(32) finalize_stats_kernel(
    const float* __restrict__ gramPart, const float* __restrict__ mmPart,
    float* __restrict__ stats, int nb, int N) {
  __shared__ double sG[25];
  int t = threadIdx.x;
  if (t < 25) {
    double g = 0.0;
    for (int b = 0; b < nb; ++b) g += (double)gramPart[b * 32 + t];
    sG[t] = g;    // t = n*5 + m  -> G[m][n]
  }
  __syncthreads();
  if (t == 0) {
    double mean[4];
    for (int j = 0; j < 4; ++j) mean[j] = sG[j * 5 + 4] / (double)N; // G[4][j]
    double cov[4][4];
    for (int i = 0; i < 4; ++i)
      for (int j = 0; j < 4; ++j) {
        double g = sG[j * 5 + i]; // G[i][j]
        cov[i][j] = (g - (double)N * mean[i] * mean[j]) / (double)(N - 1);
        if (i == j) cov[i][j] += 1e-6;
      }
    double L[4][4] = {};
    for (int i = 0; i < 4; ++i)
      for (int j = 0; j <= i; ++j) {
        double s = cov[i][j];
        for (int k = 0; k < j; ++k) s -= L[i][k] * L[j][k];
        L[i][j] = (i == j) ? sqrt(s) : s / L[j][j];
      }
    for (int j = 0; j < 4; ++j) stats[j] = (float)mean[j];
    for (int i = 0; i < 4; ++i)
      for (int j = 0; j < 4; ++j) stats[4 + i * 4 + j] = (float)L[i][j];
    float qmn = __builtin_inff(), qmx = -__builtin_inff();
    float rmn = __builtin_inff(), rmx = -__builtin_inff();
    for (int b = 0; b < nb; ++b) {
      qmn = fminf(qmn, mmPart[b * 4 + 0]);
      qmx = fmaxf(qmx, mmPart[b * 4 + 1]);
      rmn = fminf(rmn, mmPart[b * 4 + 2]);
      rmx = fmaxf(rmx, mmPart[b * 4 + 3]);
    }
    stats[20] = qmn; stats[21] = qmx; stats[22] = rmn; stats[23] = rmx;
  }
}

// ------------------------------------------------- marginal 50^2 histograms
__global__ void __launch_bounds__(256) marginal_hist_kernel(
    const float* __restrict__ q, const float* __restrict__ r,
    unsigned int* __restrict__ histT, unsigned int* __restrict__ histI,
    const float* __restrict__ stats, int N) {
  float qlo = stats[20], qhi = stats[21], rlo = stats[22], rhi = stats[23];
  float sq = (float)BINS / (qhi - qlo);
  float sr = (float)BINS / (rhi - rlo);
  int i = blockIdx.x * blockDim.x + threadIdx.x;
  int stride = gridDim.x * blockDim.x;
  for (; i < N; i += stride) {
    float a0 = q[i * 2], a1 = q[i * 2 + 1];
    int i0 = min(max((int)floorf((a0 - qlo) * sq), 0), BINS - 1);
    int i1 = min(max((int)floorf((a1 - qlo) * sq), 0), BINS - 1);
    atomicAdd(&histT[i0 * BINS + i1], 1u);
    float b0 = r[i * 2], b1 = r[i * 2 + 1];
    int j0 = min(max((int)floorf((b0 - rlo) * sr), 0), BINS - 1);
    int j1 = min(max((int)floorf((b1 - rlo) * sr), 0), BINS - 1);
    atomicAdd(&histI[j0 * BINS + j1], 1u);
  }
}

// ------------------------------------------------- sample transform helper
__device__ __forceinline__ void compute_sample(const float4 zv,
                                               const float* __restrict__ Lm,
                                               const float* __restrict__ mean,
                                               float s[4]) {
#pragma unroll
  for (int j = 0; j < 4; ++j)
    s[j] = mean[j] + zv.x * Lm[j * 4 + 0] + zv.y * Lm[j * 4 + 1] +
           zv.z * Lm[j * 4 + 2] + zv.w * Lm[j * 4 + 3];
}

// ---------------------------------------------- pass 1 over z: global min/max
__global__ void __launch_bounds__(256) samples_minmax_kernel(
    const float* __restrict__ z, const float* __restrict__ stats,
    float* __restrict__ smm, int S) {
  __shared__ float shmn[8], shmx[8];
  float mean[4], Lm[16];
#pragma unroll
  for (int j = 0; j < 4; ++j) mean[j] = stats[j];
#pragma unroll
  for (int j = 0; j < 16; ++j) Lm[j] = stats[4 + j];
  float mn = __builtin_inff(), mx = -__builtin_inff();
  const float4* __restrict__ z4 = (const float4*)z;
  int i = blockIdx.x * blockDim.x + threadIdx.x;
  int stride = gridDim.x * blockDim.x;
  for (; i < S; i += stride) {
    __builtin_prefetch(z4 + i + stride, 0, 0);
    float4 zv = z4[i];
    float s[4];
    compute_sample(zv, Lm, mean, s);
#pragma unroll
    for (int j = 0; j < 4; ++j) {
      mn = fminf(mn, s[j]);
      mx = fmaxf(mx, s[j]);
    }
  }
#pragma unroll
  for (int o = 16; o; o >>= 1) {
    mn = fminf(mn, __shfl_xor(mn, o, 32));
    mx = fmaxf(mx, __shfl_xor(mx, o, 32));
  }
  int w = threadIdx.x >> 5;
  if ((threadIdx.x & 31) == 0) { shmn[w] = mn; shmx[w] = mx; }
  __syncthreads();
  if (threadIdx.x == 0) {
    for (int k = 1; k < 8; ++k) { mn = fminf(mn, shmn[k]); mx = fmaxf(mx, shmx[k]); }
    smm[blockIdx.x * 2] = mn;
    smm[blockIdx.x * 2 + 1] = mx;
  }
}

__global__ void __launch_bounds__(256) smm_reduce_kernel(
    const float* __restrict__ smm, float* __restrict__ stats, int nb) {
  __shared__ float shmn[8], shmx[8];
  float mn = __builtin_inff(), mx = -__builtin_inff();
  for (int b = threadIdx.x; b < nb; b += blockDim.x) {
    mn = fminf(mn, smm[b * 2]);
    mx = fmaxf(mx, smm[b * 2 + 1]);
  }
#pragma unroll
  for (int o = 16; o; o >>= 1) {
    mn = fminf(mn, __shfl_xor(mn, o, 32));
    mx = fmaxf(mx, __shfl_xor(mx, o, 32));
  }
  int w = threadIdx.x >> 5;
  if ((threadIdx.x & 31) == 0) { shmn[w] = mn; shmx[w] = mx; }
  __syncthreads();
  if (threadIdx.x == 0) {
    for (int k = 1; k < 8; ++k) { mn = fminf(mn, shmn[k]); mx = fmaxf(mx, shmx[k]); }
    stats[24] = mn;
    stats[25] = mx;
  }
}

// ---------------------------------- pass 2 over z: 50^4 joint histogram (L2-resident)
__global__ void __launch_bounds__(256) joint_hist_kernel(
    const float* __restrict__ z, const float* __restrict__ stats,
    unsigned int* __restrict__ histJ, int S) {
  float mean[4], Lm[16];
#pragma unroll
  for (int j = 0; j < 4; ++j) mean[j] = stats[j];
#pragma unroll
  for (int j = 0; j < 16; ++j) Lm[j] = stats[4 + j];
  float lo = stats[24], hi = stats[25];
  float sc = (float)BINS / (hi - lo);
  const float4* __restrict__ z4 = (const float4*)z;
  int i = blockIdx.x * blockDim.x + threadIdx.x;
  int stride = gridDim.x * blockDim.x;
  for (; i < S; i += stride) {
    __builtin_prefetch(z4 + i + stride, 0, 0);
    float4 zv = z4[i];
    float s[4];
    compute_sample(zv, Lm, mean, s);
    int idx[4];
#pragma unroll
    for (int j = 0; j < 4; ++j)
      idx[j] = min(max((int)floorf((s[j] - lo) * sc), 0), BINS - 1);
    int flat = ((idx[0] * BINS + idx[1]) * BINS + idx[2]) * BINS + idx[3];
    atomicAdd(&histJ[flat], 1u);
  }
}

// ------------------------------------------------------- entropy partials
__global__ void __launch_bounds__(256) entropy_kernel(
    const unsigned int* __restrict__ hist, int nbins, float invTotal,
    float* __restrict__ part) {
  __shared__ float sh[8];
  float acc = 0.f;
  int i = blockIdx.x * blockDim.x + threadIdx.x;
  int stride = gridDim.x * blockDim.x;
  for (; i < nbins; i += stride) {
    unsigned int c = hist[i];
    if (c) {
      float p = (float)c * invTotal;
      acc -= p * logf(p);
    }
  }
#pragma unroll
  for (int o = 16; o; o >>= 1) acc += __shfl_xor(acc, o, 32);
  int w = threadIdx.x >> 5;
  if ((threadIdx.x & 31) == 0) sh[w] = acc;
  __syncthreads();
  if (threadIdx.x == 0) {
    for (int k = 1; k < 8; ++k) acc += sh[k];
    part[blockIdx.x] = acc;
  }
}

__global__ void final_kernel(const float* __restrict__ partJ, int nJ,
                             const float* __restrict__ partT, int nT,
                             const float* __restrict__ partI, int nI,
                             float* __restrict__ out) {
  if (blockIdx.x == 0 && threadIdx.x == 0) {
    double Hj = 0, Ht = 0, Hi = 0;
    for (int k = 0; k < nJ; ++k) Hj += (double)partJ[k];
    for (int k = 0; k < nT; ++k) Ht += (double)partT[k];
    for (int k = 0; k < nI; ++k) Hi += (double)partI[k];
    double v = Hj / (Ht + Hi);
    v = fmin(fmax(v, 0.0), 1.0);
    out[0] = (float)v;
  }
}

// --------------------------------------------------------------- launcher
extern "C" void kernel_launch(void* const* d_in, const int* in_sizes, int n_in,
                              void* d_out, int out_size, void* d_ws, size_t ws_size,
                              hipStream_t stream) {
  const float* q = (const float*)d_in[0];
  const float* r = (const float*)d_in[1];
  const float* z = (const float*)d_in[2];
  const int N = in_sizes[0] / 2;   // 200000 rows, D=2
  const int S = in_sizes[2] / 4;   // 2000000 rows, 2D=4

  const long B4 = (long)BINS * BINS * BINS * BINS; // 6,250,000
  const long B2 = (long)BINS * BINS;               // 2,500

  unsigned int* histJ = (unsigned int*)d_ws;
  unsigned int* histT = histJ + B4;
  unsigned int* histI = histT + B2;
  float* gramPart = (float*)(histI + B2);        // NB_COV * 32
  float* mmPart   = gramPart + NB_COV * 32;      // NB_COV * 4
  float* stats    = mmPart + NB_COV * 4;         // 32 floats: mean[4],L[16],ranges
  float* smm      = stats + 32;                  // NB_SMM * 2
  float* partJ    = smm + NB_SMM * 2;            // NB_ENTJ
  float* partT    = partJ + NB_ENTJ;             // NB_ENTM
  float* partI    = partT + NB_ENTM;             // NB_ENTM
  float* wsEnd    = partI + NB_ENTM;
  long nWords = (long)((unsigned int*)wsEnd - histJ);

  zero_ws_kernel<<<2048, 256, 0, stream>>>((unsigned int*)d_ws, nWords);
  cov_wmma_kernel<<<NB_COV, 32, 0, stream>>>(q, r, gramPart, mmPart, N / 4);
  finalize_stats_kernel<<<1, 32, 0, stream>>>(gramPart, mmPart, stats, NB_COV, N);
  marginal_hist_kernel<<<(N + 255) / 256, 256, 0, stream>>>(q, r, histT, histI, stats, N);
  samples_minmax_kernel<<<NB_SMM, 256, 0, stream>>>(z, stats, smm, S);
  smm_reduce_kernel<<<1, 256, 0, stream>>>(smm, stats, NB_SMM);
  joint_hist_kernel<<<NB_HISTJ, 256, 0, stream>>>(z, stats, histJ, S);
  entropy_kernel<<<NB_ENTJ, 256, 0, stream>>>(histJ, (int)B4, 1.0f / (float)S, partJ);
  entropy_kernel<<<NB_ENTM, 256, 0, stream>>>(histT, (int)B2, 1.0f / (float)N, partT);
  entropy_kernel<<<NB_ENTM, 256, 0, stream>>>(histI, (int)B2, 1.0f / (float)N, partI);
  final_kernel<<<1, 32, 0, stream>>>(partJ, NB_ENTJ, partT, NB_ENTM, partI, NB_ENTM,
                                     (float*)d_out);
}